// SGCRes_10316511445629
// MI455X (gfx1250) — compile-verified
//
#include <hip/hip_runtime.h>
#include <hip/hip_bf16.h>

typedef __attribute__((ext_vector_type(2))) float v2f;
typedef __attribute__((ext_vector_type(8))) float v8f;

#define FEAT_D 128
#define LN_EPS 1e-5f

// ---------------------------------------------------------------------------
// Zero workspace (atomic accumulators must start at 0 each launch).
// ---------------------------------------------------------------------------
__global__ void __launch_bounds__(256) zero_f4(float4* __restrict__ p, long n4) {
    long i = (long)blockIdx.x * blockDim.x + threadIdx.x;
    if (i < n4) {
        float4 z; z.x = 0.f; z.y = 0.f; z.z = 0.f; z.w = 0.f;
        p[i] = z;
    }
}

// ---------------------------------------------------------------------------
// SpMM: out[dst] += h[src], unit edge weights. One wave32 per edge:
// each lane handles 4 of the 128 feature floats with stride-32 layout so
// every load and every global_atomic_add_f32 instruction is a fully
// coalesced 128B transaction. Both h and out are L2-resident (51.2MB each
// vs 192MB L2), so this runs at L2/atomic throughput, not HBM.
// ---------------------------------------------------------------------------
__global__ void __launch_bounds__(256) spmm_edges(const float* __restrict__ h,
                                                  const int* __restrict__ src,
                                                  const int* __restrict__ dst,
                                                  float* __restrict__ out,
                                                  int E) {
    int edge = blockIdx.x * 8 + (threadIdx.x >> 5);   // 8 waves per block
    if (edge >= E) return;
    int lane = threadIdx.x & 31;
    int s = src[edge];
    int d = dst[edge];
    const float* hs = h + (size_t)s * FEAT_D;
    float* od = out + (size_t)d * FEAT_D;
#pragma unroll
    for (int i = 0; i < 4; ++i) {
        int c = lane + 32 * i;
        atomicAdd(&od[c], hs[c]);
    }
}

// ---------------------------------------------------------------------------
// Fused GEMM (h2 @ W^T + b) + LayerNorm + affine, using V_WMMA_F32_16X16X4_F32.
//
// Block = 256 threads = 8 waves, owns 16 output rows.
//   - stage 16x128 h2 tile in LDS once (shared by all 8 waves)
//   - wave w computes the 16x16 tile for classes [16w, 16w+16)
//   - K=128 swept in 32 steps of 4 via f32 WMMA
//   - A frag (16x4): lanes 0-15 hold K={k,k+1} of row M=lane, lanes 16-31
//     hold K={k+2,k+3}  -> one ds_load_b64 per lane per step
//   - B frag (4x16):  B[k'][n] = W[c0+n][k'], same pairing -> one
//     global_load_b64 per lane per step (W is 64KB, L2 resident)
//   - D layout: acc[j] = D[M = j + 8*(lane>=16)][N = c0 + lane%16]
// Then tiles go through LDS for the cross-wave row mean/var reduction and
// the normalized result is written to global.
// ---------------------------------------------------------------------------
#define SD_STRIDE 132

__global__ void __launch_bounds__(256) gemm_ln(const float* __restrict__ h2,
                                               const float* __restrict__ Wm,
                                               const float* __restrict__ bias,
                                               const float* __restrict__ gamma,
                                               const float* __restrict__ beta,
                                               float* __restrict__ out,
                                               int N) {
    __shared__ float sA[16 * FEAT_D];        // h2 tile
    __shared__ float sD[16 * SD_STRIDE];     // output tile (padded stride)
    __shared__ float redS[16][17];
    __shared__ float redQ[16][17];
    __shared__ float muS[16];
    __shared__ float rsS[16];

    const int row0 = blockIdx.x * 16;
    const int t = threadIdx.x;

    // ---- stage 16x128 h2 tile: thread t -> row t>>4, 8-float chunk t&15 ----
    {
        int r = t >> 4;
        int chunk = t & 15;
        int grow = row0 + r;
        if (grow > N - 1) grow = N - 1;           // clamp (N%16==0 in practice)
        const float4* gp = (const float4*)(h2 + (size_t)grow * FEAT_D + chunk * 8);
        float4* lp = (float4*)(&sA[r * FEAT_D + chunk * 8]);
        lp[0] = gp[0];
        lp[1] = gp[1];
    }
    __syncthreads();

    const int wave = t >> 5;        // 0..7
    const int lane = t & 31;
    const int c0 = wave * 16;
    const int halfsel = lane >> 4;  // 0: K pair {k,k+1}; 1: {k+2,k+3}
    const int l16 = lane & 15;

    const float* wrow = Wm + (size_t)(c0 + l16) * FEAT_D + 2 * halfsel;
    const float* arow = &sA[l16 * FEAT_D + 2 * halfsel];

    v8f acc = {};
#pragma unroll
    for (int k = 0; k < FEAT_D; k += 4) {
        v2f a = *(const v2f*)(arow + k);    // ds_load_b64 from LDS tile
        v2f b = *(const v2f*)(wrow + k);    // global_load_b64, L2-resident W
        acc = __builtin_amdgcn_wmma_f32_16x16x4_f32(
            /*neg_a=*/false, a, /*neg_b=*/false, b,
            /*c_mod=*/(short)0, acc, /*reuse_a=*/false, /*reuse_b=*/false);
    }

    // ---- add bias, park tile in LDS for the row reduction ----
    {
        float bv = bias[c0 + l16];
#pragma unroll
        for (int j = 0; j < 8; ++j) {
            int m = j + 8 * halfsel;                     // local row 0..15
            sD[m * SD_STRIDE + c0 + l16] = acc[j] + bv;
        }
    }
    __syncthreads();

    // ---- LayerNorm: thread t -> row t>>4, 8-col group t&15 ----
    const int row = t >> 4;
    const int grp = t & 15;
    {
        float s = 0.f, q = 0.f;
#pragma unroll
        for (int j = 0; j < 8; ++j) {
            float v = sD[row * SD_STRIDE + grp * 8 + j];
            s += v;
            q += v * v;
        }
        redS[row][grp] = s;
        redQ[row][grp] = q;
    }
    __syncthreads();
    if (t < 16) {
        float S = 0.f, Q = 0.f;
#pragma unroll
        for (int g = 0; g < 16; ++g) { S += redS[t][g]; Q += redQ[t][g]; }
        float mu = S * (1.0f / FEAT_D);
        float var = Q * (1.0f / FEAT_D) - mu * mu;
        muS[t] = mu;
        rsS[t] = rsqrtf(var + LN_EPS);
    }
    __syncthreads();

    if (row0 + row < N) {
        float mu = muS[row];
        float rs = rsS[row];
        float* orow = out + (size_t)(row0 + row) * FEAT_D;
#pragma unroll
        for (int j = 0; j < 8; ++j) {
            int col = grp * 8 + j;
            float v = sD[row * SD_STRIDE + col];
            orow[col] = gamma[col] * ((v - mu) * rs) + beta[col];
        }
    }
}

// ---------------------------------------------------------------------------
// Launch: zero scratch -> SpMM hop1 -> SpMM hop2 -> fused GEMM+LN.
// ---------------------------------------------------------------------------
extern "C" void kernel_launch(void* const* d_in, const int* in_sizes, int n_in,
                              void* d_out, int out_size, void* d_ws, size_t ws_size,
                              hipStream_t stream) {
    const float* feat  = (const float*)d_in[0];
    // d_in[1] = feat_ori (dead code in reference)
    const float* Wm    = (const float*)d_in[2];
    const float* bias  = (const float*)d_in[3];
    const float* gamma = (const float*)d_in[4];
    const float* beta  = (const float*)d_in[5];
    const int*   src   = (const int*)d_in[6];
    const int*   dst   = (const int*)d_in[7];

    const int N = in_sizes[0] / FEAT_D;
    const int E = in_sizes[6];

    float* tmp1 = (float*)d_ws;
    float* tmp2 = tmp1 + (size_t)N * FEAT_D;

    // zero both accumulator buffers (2*N*128 floats = N*64 float4)
    long nz = (long)N * 64;
    zero_f4<<<(int)((nz + 255) / 256), 256, 0, stream>>>((float4*)d_ws, nz);

    int spmm_blocks = (E + 7) / 8;   // wave32 per edge, 8 waves per block
    spmm_edges<<<spmm_blocks, 256, 0, stream>>>(feat, src, dst, tmp1, E);
    spmm_edges<<<spmm_blocks, 256, 0, stream>>>(tmp1, src, dst, tmp2, E);

    int gemm_blocks = (N + 15) / 16;
    gemm_ln<<<gemm_blocks, 256, 0, stream>>>(tmp2, Wm, bias, gamma, beta,
                                             (float*)d_out, N);
}